// Points_89189290869071
// MI455X (gfx1250) — compile-verified
//
#include <hip/hip_runtime.h>
#include <hip/hip_bf16.h>

typedef float v4f __attribute__((ext_vector_type(4)));

// Problem constants (from the reference): B=8, L=2048, D=512, DELTA=3, FI=-1
constexpr int kDelta = 3;
constexpr int kW     = 2 * kDelta + 1;     // 7
constexpr int kSeq   = 2048;               // L
constexpr int kDim   = 512;                // D
constexpr int kDim4  = kDim / 4;           // 128 float4 per row
constexpr int kTile  = 16;                 // positions per block
constexpr int kRows  = kTile + 2 * kDelta; // 22 staged rows (45056 B LDS)

__global__ __launch_bounds__(256)
void gather_neighbors_kernel(const float* __restrict__ x, float* __restrict__ out) {
  __shared__ __align__(16) float lds[kRows * kDim];

  const int tid = threadIdx.x;
  const int blk = blockIdx.x;            // b * (L/kTile) + tile
  const int b   = blk >> 7;              // L/kTile == 128
  const int i0  = (blk & 127) * kTile;
  const int rlo = i0 - kDelta;           // window base (may be negative)

  const int firstRow = rlo < 0 ? 0 : rlo;
  int lastRow = i0 + kTile - 1 + kDelta;
  if (lastRow > kSeq - 1) lastRow = kSeq - 1;
  const int nRows = lastRow - firstRow + 1;

  const float* xb = x + (size_t)b * kSeq * kDim;

  // blockDim 256 == 2*kDim4: d4 is loop-invariant per thread, row counter steps by 2
  const int d4 = tid & (kDim4 - 1);
  const int r0 = tid >> 7;               // 0 or 1

  // ---- Phase 1: async-stage rows [firstRow, lastRow] into LDS (CDNA5 async copy) ----
  for (int r = r0; r < nRows; r += 2) {
    const int grow = firstRow + r;       // global row
    const int lrow = grow - rlo;         // LDS slot (0..21)
    // LDS byte address = low 32 bits of the generic pointer (aperture: addr[31:0])
    unsigned int lds_addr =
        (unsigned int)(size_t)(const void*)(lds + lrow * kDim + 4 * d4);
    unsigned long long gaddr =
        (unsigned long long)(size_t)(const void*)(xb + (size_t)grow * kDim + 4 * d4);
    // GLOBAL_LOAD_ASYNC_TO_LDS_B128, GV mode (vdst = LDS addr, v[addr], saddr=off)
    asm volatile("global_load_async_to_lds_b128 %0, %1, off"
                 :
                 : "v"(lds_addr), "v"(gaddr)
                 : "memory");
  }

  // Wait for the async copy engine (ASYNCcnt) before any lane reads LDS.
  asm volatile("s_wait_asynccnt 0x0" ::: "memory");
  __syncthreads();

  // ---- Phase 2: fan out 7 neighbor copies per position, NT-stream to HBM ----
  float* ob = out + (size_t)(b * kSeq + i0) * (kW * kDim);
  for (int rc = r0; rc < kTile * kW; rc += 2) {   // 112 row-copies per tile
    const int w  = rc % kW;
    const int li = rc / kW;
    const int i  = i0 + li;
    // neighbor slot -> source row (matches the reference's unrolled j-order + self pad)
    const int nl = i < kDelta ? i : kDelta;                           // valid left neighbors
    const int nr = (kSeq - 1 - i) < kDelta ? (kSeq - 1 - i) : kDelta; // valid right neighbors
    int src;
    if (w < nl)           src = i + w - nl;       // j in [-nl, -1]
    else if (w < nl + nr) src = i + w - nl + 1;   // j in [1, nr]
    else                  src = i;                // self padding
    const int lrow = src - rlo;                   // always within staged window
    v4f val = *((const v4f*)(lds + lrow * kDim) + d4);                // ds_load_b128
    __builtin_nontemporal_store(val, (v4f*)ob + (size_t)rc * kDim4 + d4); // b128 NT store
  }
}

extern "C" void kernel_launch(void* const* d_in, const int* in_sizes, int n_in,
                              void* d_out, int out_size, void* d_ws, size_t ws_size,
                              hipStream_t stream) {
  const float* x = (const float*)d_in[0];
  float* out = (float*)d_out;
  (void)in_sizes; (void)n_in; (void)out_size; (void)d_ws; (void)ws_size;

  dim3 grid(8 * (kSeq / kTile));   // 8 batches * 128 tiles = 1024 blocks
  dim3 block(256);                 // 8 wave32
  hipLaunchKernelGGL(gather_neighbors_kernel, grid, block, 0, stream, x, out);
}